// TorchJSDBaseline_29892972380471
// MI455X (gfx1250) — compile-verified
//
#include <hip/hip_runtime.h>
#include <hip/hip_bf16.h>

#define IGNORE_INDEX (-100)
#define LN2F 0.69314718055994530942f

// Native clang vector type: accepted by __builtin_nontemporal_load and lowers
// to a single 128-bit global load with the NT temporal hint.
typedef float vfloat4 __attribute__((ext_vector_type(4)));

// Per-element fused JSD term (beta = 0.5):
//   p = e^lp, q = e^lq, m = (p+q)/2, log m = log(p+q) - ln2
//   term = 0.5 * ( p*(lp - logm) + q*(lq - logm) )
__device__ __forceinline__ float jsd_term(float lq, float lp) {
    float p    = __expf(lp);
    float q    = __expf(lq);
    float logm = __logf(p + q) - LN2F;
    return 0.5f * (p * (lp - logm) + q * (lq - logm));
}

// One block per row. 256 threads = 8 wave32 waves.
// Streams 2 * vocab fp32 with non-temporal 128-bit loads, reduces in LDS,
// one global_atomic_add_f32 per (valid) row.
__global__ __launch_bounds__(256) void jsd_row_kernel(
    const float* __restrict__ log_q,
    const float* __restrict__ log_p,
    const int*   __restrict__ label,
    float*       __restrict__ sum_acc,
    int vocab)
{
    const int row = blockIdx.x;
    const int tid = threadIdx.x;

    // label[] is uniform per block: every thread takes the same branch.
    if (label[row] == IGNORE_INDEX) return;

    const size_t base = (size_t)row * (size_t)vocab;
    const vfloat4* q4 = reinterpret_cast<const vfloat4*>(log_q + base) + tid;
    const vfloat4* p4 = reinterpret_cast<const vfloat4*>(log_p + base) + tid;
    const int n4 = vocab >> 2;

    // Wave-uniform main loop: every lane runs exactly `iters` iterations
    // (scalar loop control, no EXEC divergence). Remainder handled below.
    const int iters = n4 >> 8;          // full strips of 256 vec4
    const int strip = iters << 8;       // vec4s covered by the uniform loop

    // 4 independent accumulators break the serial FP dependency chain so the
    // exp/log pipelines stay busy across the unrolled iterations.
    float a0 = 0.0f, a1 = 0.0f, a2 = 0.0f, a3 = 0.0f;

#pragma unroll 4
    for (int it = 0; it < iters; ++it) {
        vfloat4 lq = __builtin_nontemporal_load(q4 + (size_t)it * 256);
        vfloat4 lp = __builtin_nontemporal_load(p4 + (size_t)it * 256);
        a0 += jsd_term(lq.x, lp.x);
        a1 += jsd_term(lq.y, lp.y);
        a2 += jsd_term(lq.z, lp.z);
        a3 += jsd_term(lq.w, lp.w);
    }

    // Partial strip: one predicated vec4 for lanes with strip+tid < n4.
    if (strip + tid < n4) {
        vfloat4 lq = __builtin_nontemporal_load(q4 + (size_t)strip);
        vfloat4 lp = __builtin_nontemporal_load(p4 + (size_t)strip);
        a0 += jsd_term(lq.x, lp.x);
        a1 += jsd_term(lq.y, lp.y);
        a2 += jsd_term(lq.z, lp.z);
        a3 += jsd_term(lq.w, lp.w);
    }

    float acc = (a0 + a1) + (a2 + a3);

    // Scalar tail (vocab not multiple of 4) — not taken for VOCAB=32000.
    for (int i = (n4 << 2) + tid; i < vocab; i += 256) {
        float lq = __builtin_nontemporal_load(log_q + base + i);
        float lp = __builtin_nontemporal_load(log_p + base + i);
        acc += jsd_term(lq, lp);
    }

    // Block reduction in LDS (wave32: 8 waves -> tree over 256 slots).
    __shared__ float red[256];
    red[tid] = acc;
    __syncthreads();
#pragma unroll
    for (int s = 128; s > 0; s >>= 1) {
        if (tid < s) red[tid] += red[tid + s];
        __syncthreads();
    }

    if (tid == 0) {
        atomicAdd(sum_acc, red[0]);   // global_atomic_add_f32, no return needed
    }
}

// Single-block finalize: count valid labels, divide the accumulated sum.
__global__ __launch_bounds__(256) void jsd_finalize_kernel(
    const float* __restrict__ sum_acc,
    const int*   __restrict__ label,
    int n_rows,
    float*       __restrict__ out)
{
    __shared__ int cnt[256];
    const int tid = threadIdx.x;

    int c = 0;
    for (int i = tid; i < n_rows; i += 256)
        c += (label[i] != IGNORE_INDEX) ? 1 : 0;

    cnt[tid] = c;
    __syncthreads();
#pragma unroll
    for (int s = 128; s > 0; s >>= 1) {
        if (tid < s) cnt[tid] += cnt[tid + s];
        __syncthreads();
    }

    if (tid == 0) {
        float n = (float)cnt[0];
        out[0] = (n > 0.0f) ? (sum_acc[0] / fmaxf(n, 1.0f)) : 0.0f;
    }
}

extern "C" void kernel_launch(void* const* d_in, const int* in_sizes, int n_in,
                              void* d_out, int out_size, void* d_ws, size_t ws_size,
                              hipStream_t stream) {
    const float* log_q = (const float*)d_in[0];
    const float* log_p = (const float*)d_in[1];
    const int*   label = (const int*)d_in[2];
    float*       out   = (float*)d_out;

    const int n_rows = in_sizes[2];
    const int vocab  = (n_rows > 0) ? (in_sizes[0] / n_rows) : 0;

    float* sum_acc = (float*)d_ws;

    // Zero the accumulator every call (deterministic; capturable memset node).
    (void)hipMemsetAsync(sum_acc, 0, sizeof(float), stream);

    jsd_row_kernel<<<n_rows, 256, 0, stream>>>(log_q, log_p, label, sum_acc, vocab);
    jsd_finalize_kernel<<<1, 256, 0, stream>>>(sum_acc, label, n_rows, out);
}